// GCN_89764816486749
// MI455X (gfx1250) — compile-verified
//
#include <hip/hip_runtime.h>

// ---------------------------------------------------------------------------
// GCN (2x GCNConv) for MI455X / gfx1250.
//  - fp32 WMMA (v_wmma_f32_16x16x4_f32) for the dense  x @ W1  GEMM
//  - float global atomics for the segment-sum scatters (L2-resident)
// Inputs (setup_inputs order):
//   d_in[0] x          [65536,128] f32
//   d_in[1] W1         [128,16]    f32
//   d_in[2] b1         [16]        f32
//   d_in[3] W2         [16,1]      f32
//   d_in[4] b2         [1]         f32
//   d_in[5] edge_index [2,2097152] i64
//   d_in[6] batch      scalar (unused)
// Output: [64,1024] f32 == 65536 floats flat.
// ---------------------------------------------------------------------------

typedef __attribute__((ext_vector_type(2))) float v2f;
typedef __attribute__((ext_vector_type(8))) float v8f;

#define NFEAT 128
#define NLAT  16

// deg[i] = 1 (self loop)
__global__ __launch_bounds__(256) void gcn_deg_init(float* __restrict__ deg, int n) {
  int i = blockIdx.x * 256 + threadIdx.x;
  if (i < n) deg[i] = 1.0f;
}

// deg[dst[e]] += 1
__global__ __launch_bounds__(256) void gcn_deg_accum(const long long* __restrict__ dst,
                                                     float* __restrict__ deg, int E) {
  int e = blockIdx.x * 256 + threadIdx.x;
  if (e < E) atomicAdd(&deg[(int)dst[e]], 1.0f);
}

// deg -> deg^-1/2 in place (deg >= 1 always, self loops)
__global__ __launch_bounds__(256) void gcn_dinv(float* __restrict__ deg, int n) {
  int i = blockIdx.x * 256 + threadIdx.x;
  if (i < n) deg[i] = rsqrtf(deg[i]);
}

// h = x @ W1  via V_WMMA_F32_16X16X4_F32.
// Block = 256 threads = 8 wave32, each wave computes one 16x16 tile of h.
// W1 (128x16 = 8KB) staged in LDS once per block.
// Fragment layouts per CDNA5 ISA 7.12.2 (fp32):
//   A 16x4 : lanes 0-15 -> row M=lane, VGPR0/1 = K=0,1 ; lanes 16-31 -> K=2,3
//   B 4x16 : lane n = col; VGPR0/1 = rows K=khalf, khalf+1 (khalf = 2*(lane>=16))
//   C/D    : VGPR r -> row (r + 8*(lane>=16)), col = lane&15
__global__ __launch_bounds__(256) void gcn_gemm1_wmma(const float* __restrict__ x,
                                                      const float* __restrict__ W1,
                                                      float* __restrict__ h) {
  __shared__ float sW[NFEAT * NLAT];
  for (int i = threadIdx.x; i < NFEAT * NLAT; i += 256) sW[i] = W1[i];
  __syncthreads();

  const int wave  = threadIdx.x >> 5;
  const int lane  = threadIdx.x & 31;
  const int n     = lane & 15;          // B col / A row within tile
  const int khalf = (lane >> 4) << 1;   // 0 or 2
  const size_t row0 = ((size_t)blockIdx.x * 8 + wave) * 16;

  const float* arow = x + (row0 + (size_t)n) * NFEAT + khalf;

  v8f acc = {};
#pragma unroll
  for (int k0 = 0; k0 < NFEAT; k0 += 4) {
    v2f a, b;
    a.x = arow[k0];
    a.y = arow[k0 + 1];
    b.x = sW[(k0 + khalf) * NLAT + n];
    b.y = sW[(k0 + khalf + 1) * NLAT + n];
    // (neg_a, A, neg_b, B, c_mod, C, reuse_a, reuse_b)
    acc = __builtin_amdgcn_wmma_f32_16x16x4_f32(false, a, false, b, (short)0, acc,
                                                false, false);
  }

  const int mbase = (lane >> 4) << 3;   // 0 or 8
#pragma unroll
  for (int r = 0; r < 8; ++r)
    h[(row0 + (size_t)(mbase + r)) * NLAT + n] = acc[r];
}

// out1 = self-loop term: dinv[i]^2 * h[i][c]
__global__ __launch_bounds__(256) void gcn_self1(const float* __restrict__ dinv,
                                                 const float* __restrict__ h,
                                                 float* __restrict__ out1, int nElem) {
  int t = blockIdx.x * 256 + threadIdx.x;
  if (t < nElem) {
    float di = dinv[t >> 4];
    out1[t] = di * di * h[t];
  }
}

// out1[dst] += dinv[src]*dinv[dst] * h[src]   (16 channels / edge, 64B aligned)
__global__ __launch_bounds__(256) void gcn_agg1(const long long* __restrict__ src,
                                                const long long* __restrict__ dst,
                                                const float* __restrict__ dinv,
                                                const float* __restrict__ h,
                                                float* __restrict__ out1, int E) {
  int e = blockIdx.x * 256 + threadIdx.x;
  if (e >= E) return;
  int s = (int)src[e];
  int d = (int)dst[e];
  float w = dinv[s] * dinv[d];
  const float4* hs = (const float4*)(h + (size_t)s * NLAT);
  float* od = out1 + (size_t)d * NLAT;
#pragma unroll
  for (int q = 0; q < 4; ++q) {
    float4 v = hs[q];
    atomicAdd(od + 4 * q + 0, w * v.x);
    atomicAdd(od + 4 * q + 1, w * v.y);
    atomicAdd(od + 4 * q + 2, w * v.z);
    atomicAdd(od + 4 * q + 3, w * v.w);
  }
}

// h2[i] = relu(out1[i] + b1) . W2      (16-wide dot per node)
__global__ __launch_bounds__(256) void gcn_relu_gemm2(const float* __restrict__ out1,
                                                      const float* __restrict__ b1,
                                                      const float* __restrict__ W2,
                                                      float* __restrict__ h2, int n) {
  int i = blockIdx.x * 256 + threadIdx.x;
  if (i >= n) return;
  const float4* r = (const float4*)(out1 + (size_t)i * NLAT);
  float s = 0.f;
#pragma unroll
  for (int q = 0; q < 4; ++q) {
    float4 v = r[q];
    float a0 = v.x + b1[4 * q + 0]; a0 = a0 > 0.f ? a0 : 0.f;
    float a1 = v.y + b1[4 * q + 1]; a1 = a1 > 0.f ? a1 : 0.f;
    float a2 = v.z + b1[4 * q + 2]; a2 = a2 > 0.f ? a2 : 0.f;
    float a3 = v.w + b1[4 * q + 3]; a3 = a3 > 0.f ? a3 : 0.f;
    s += a0 * W2[4 * q + 0] + a1 * W2[4 * q + 1] + a2 * W2[4 * q + 2] + a3 * W2[4 * q + 3];
  }
  h2[i] = s;
}

// out[i] = dinv[i]^2 * h2[i] + b2   (initializes every element of d_out)
__global__ __launch_bounds__(256) void gcn_self2(const float* __restrict__ dinv,
                                                 const float* __restrict__ h2,
                                                 const float* __restrict__ b2,
                                                 float* __restrict__ out, int n) {
  int i = blockIdx.x * 256 + threadIdx.x;
  if (i < n) {
    float di = dinv[i];
    out[i] = di * di * h2[i] + b2[0];
  }
}

// out[dst] += dinv[src]*dinv[dst] * h2[src]
__global__ __launch_bounds__(256) void gcn_agg2(const long long* __restrict__ src,
                                                const long long* __restrict__ dst,
                                                const float* __restrict__ dinv,
                                                const float* __restrict__ h2,
                                                float* __restrict__ out, int E) {
  int e = blockIdx.x * 256 + threadIdx.x;
  if (e >= E) return;
  int s = (int)src[e];
  int d = (int)dst[e];
  atomicAdd(out + d, dinv[s] * dinv[d] * h2[s]);
}

extern "C" void kernel_launch(void* const* d_in, const int* in_sizes, int n_in,
                              void* d_out, int out_size, void* d_ws, size_t ws_size,
                              hipStream_t stream) {
  const float* x  = (const float*)d_in[0];
  const float* W1 = (const float*)d_in[1];
  const float* b1 = (const float*)d_in[2];
  const float* W2 = (const float*)d_in[3];
  const float* b2 = (const float*)d_in[4];
  const long long* ei = (const long long*)d_in[5];

  const int N = in_sizes[0] / NFEAT;        // 65536
  const int E = in_sizes[5] / 2;            // 2097152
  const long long* src = ei;
  const long long* dst = ei + E;

  // workspace layout (floats): dinv[N] | h[N*16] | out1[N*16] | h2[N]
  float* dinv = (float*)d_ws;
  float* h    = dinv + N;
  float* out1 = h + (size_t)N * NLAT;
  float* h2   = out1 + (size_t)N * NLAT;
  float* out  = (float*)d_out;

  const dim3 B(256);
  const int gN   = (N + 255) / 256;
  const int gE   = (E + 255) / 256;
  const int gNC  = (N * NLAT + 255) / 256;

  gcn_deg_init  <<<gN,      B, 0, stream>>>(dinv, N);
  gcn_deg_accum <<<gE,      B, 0, stream>>>(dst, dinv, E);
  gcn_dinv      <<<gN,      B, 0, stream>>>(dinv, N);
  gcn_gemm1_wmma<<<N / 128, B, 0, stream>>>(x, W1, h);
  gcn_self1     <<<gNC,     B, 0, stream>>>(dinv, h, out1, N * NLAT);
  gcn_agg1      <<<gE,      B, 0, stream>>>(src, dst, dinv, h, out1, E);
  gcn_relu_gemm2<<<gN,      B, 0, stream>>>(out1, b1, W2, h2, N);
  gcn_self2     <<<gN,      B, 0, stream>>>(dinv, h2, b2, out, N);
  gcn_agg2      <<<gE,      B, 0, stream>>>(src, dst, dinv, h2, out, E);
}